// CausalConvTranspose1d_86414741996179
// MI455X (gfx1250) — compile-verified
//
#include <hip/hip_runtime.h>

typedef float v2f __attribute__((ext_vector_type(2)));
typedef float v4f __attribute__((ext_vector_type(4)));
typedef float v8f __attribute__((ext_vector_type(8)));

#define NB    8
#define CIN   512
#define COUT  256
#define TIN   4096
#define KW    16
#define TOUT  (TIN * 8)          // 32768 outputs per (b,o)

#define GROUP_PAD 20             // 16 floats + 4 pad per column group
#define WAVE_LDS  (16 * GROUP_PAD)

__global__ __launch_bounds__(256)
void causal_convtr1d_wmma(const float* __restrict__ x,
                          const float* __restrict__ w,
                          const float* __restrict__ bias,
                          float* __restrict__ y)
{
    __shared__ float lds[8 * WAVE_LDS];

    const int o    = blockIdx.x;         // output channel
    const int b    = blockIdx.y;         // batch
    const int tid  = threadIdx.x;
    const int wave = tid >> 5;
    const int L    = tid & 31;           // lane
    const int p    = L & 15;
    const bool lo_half = (L < 16);

    const float* x0 = x + ((size_t)b * CIN + 2 * o) * TIN;   // channel 2o
    const float* x1 = x0 + TIN;                              // channel 2o+1
    const float* w0 = w + (size_t)(2 * o) * KW;
    const float* w1 = w0 + KW;
    float*       yo = y + ((size_t)b * COUT + o) * (size_t)TOUT;

    // ---- A matrices (16x4 f32 => v2f per lane), once per block ----
    // lanes 0-15: M=p,{K0,K1}; lanes 16-31: M=p,{K2,K3}
    // A[p][0]=(p<8)?w[p+8]:0  A[p][1]=w[p]  A[p][2]=(p>=8)?w[p-8]:0  A[p][3]=0
    const int  i0   = (lo_half ? (p + 8) : (p - 8)) & 15;
    const bool use0 = lo_half ? (p < 8) : (p >= 8);
    v2f a1, a2;
    a1.x = use0 ? w0[i0] : 0.0f;
    a1.y = lo_half ? w0[p] : 0.0f;
    a2.x = use0 ? w1[i0] : 0.0f;
    a2.y = lo_half ? w1[p] : 0.0f;

    const float bv = bias[o];
    v8f cbias;
#pragma unroll
    for (int i = 0; i < 8; ++i) cbias[i] = bv;

    float* wlds = lds + wave * WAVE_LDS;

    const int dml = lo_half ? -1 : 1;

    // ---- Preload tile 0 (unconditional loads, clamp + select for ml<0) ----
    // B layout: v0 lanes0-15 = x[2q-1], lanes16-31 = x[2q+1]
    //           v1 lanes0-15 = x[2q],   lanes16-31 = 0
    const int qq0  = wave * 16 + p;
    const int ml0  = 2 * qq0 + dml;              // -1 only for wave0/lane0
    const int ml0c = (ml0 < 0) ? 0 : ml0;
    float l0 = x0[ml0c];  l0 = (ml0 >= 0) ? l0 : 0.0f;
    float l1 = x1[ml0c];  l1 = (ml0 >= 0) ? l1 : 0.0f;
    float h0 = x0[2 * qq0];  h0 = lo_half ? h0 : 0.0f;
    float h1 = x1[2 * qq0];  h1 = lo_half ? h1 : 0.0f;

    // 128 tiles of 256 outputs per (b,o); 8 waves -> 16 tiles per wave
    for (int it = 0; it < 16; ++it) {
        const int tile = it * 8 + wave;
        const int qq   = tile * 16 + p;          // this lane's column q
        const int t0   = tile * 256;             // tile's base output t

        // Speculative GL2 prefetch two tiles ahead (q advances 128/iter)
        const int pf = (2 * qq + 512 < TIN) ? (2 * qq + 512) : (TIN - 1);
        __builtin_prefetch(x0 + pf, 0, 1);
        __builtin_prefetch(x1 + pf, 0, 1);

        // ---- Issue next iteration's B loads now (software pipeline) ----
        // qq_next = qq+128 => indices always >= 0; clamp top so the dead
        // loads of the final iteration can't fault.
        const int qn  = qq + 128;
        int mn  = 2 * qn + dml;        if (mn  > TIN - 1) mn  = TIN - 1;
        int mhn = 2 * qn;              if (mhn > TIN - 1) mhn = TIN - 1;
        float nl0 = x0[mn];
        float nl1 = x1[mn];
        float nh0 = x0[mhn];  nh0 = lo_half ? nh0 : 0.0f;
        float nh1 = x1[mhn];  nh1 = lo_half ? nh1 : 0.0f;

        // ---- Two chained WMMAs: D = A2*B2 + (A1*B1 + bias) ----
        v2f b1; b1.x = l0; b1.y = h0;
        v2f b2; b2.x = l1; b2.y = h1;

        v8f acc = __builtin_amdgcn_wmma_f32_16x16x4_f32(
            false, a1, false, b1, (short)0, cbias, false, false);
        acc = __builtin_amdgcn_wmma_f32_16x16x4_f32(
            false, a2, false, b2, (short)0, acc, false, false);

        // ---- Epilogue: per-wave LDS transpose, coalesced NT b128 stores ----
        // D: VGPR v, lanes0-15 -> (M=v, N=p); lanes16-31 -> (M=v+8, N=p)
#pragma unroll
        for (int v = 0; v < 8; ++v) {
            const int M = lo_half ? v : (v + 8);
            wlds[p * GROUP_PAD + M] = acc[v];
        }
#pragma unroll
        for (int i = 0; i < 2; ++i) {
            const int t = i * 128 + L * 4;
            const int g = t >> 4;
            const int r = t & 15;
            const v4f vv = *(const v4f*)(wlds + g * GROUP_PAD + r);
            __builtin_nontemporal_store(vv, (v4f*)(yo + t0 + t));
        }

        l0 = nl0; l1 = nl1; h0 = nh0; h1 = nh1;
    }
}

extern "C" void kernel_launch(void* const* d_in, const int* in_sizes, int n_in,
                              void* d_out, int out_size, void* d_ws, size_t ws_size,
                              hipStream_t stream) {
    (void)in_sizes; (void)n_in; (void)out_size; (void)d_ws; (void)ws_size;
    const float* x    = (const float*)d_in[0];   // (8, 512, 4096) f32
    const float* wgt  = (const float*)d_in[1];   // (512, 16, 1)   f32
    const float* bias = (const float*)d_in[2];   // (256,)         f32
    float* y = (float*)d_out;                    // (8, 256, 32768) f32

    dim3 grid(COUT, NB);
    causal_convtr1d_wmma<<<grid, 256, 0, stream>>>(x, wgt, bias, y);
}